// CCHLoss_50337016709911
// MI455X (gfx1250) — compile-verified
//
#include <hip/hip_runtime.h>

typedef __attribute__((ext_vector_type(2))) float v2f;
typedef __attribute__((ext_vector_type(8))) float v8f;

#define BATCH 8
#define NPTS 16384              // query points per batch (v_pred)
#define MPTS 16384              // target points per batch (v)
#define QT 4                    // query tiles of 16 held per wave
#define MSEG 8                  // segments over target dimension
#define TILES_PER_SEG ((MPTS / 16) / MSEG)                  // 128
#define WAVES_PER_BATCH (((NPTS / 16) / QT) * MSEG)         // 2048
#define TOTAL_WAVES (BATCH * WAVES_PER_BATCH)               // 16384
#define BLOCK_THREADS 256
#define WAVES_PER_BLOCK (BLOCK_THREADS / 32)
#define MAIN_BLOCKS (TOTAL_WAVES / WAVES_PER_BLOCK)         // 2048

// Raw CDNA5 min ops: avoid clang's canonicalize (v_max x,x) spam around fminf.
__device__ __forceinline__ float min3f(float a, float b, float c) {
    float d;
    asm("v_min3_num_f32 %0, %1, %2, %3" : "=v"(d) : "v"(a), "v"(b), "v"(c));
    return d;
}
__device__ __forceinline__ float min2f(float a, float b) {
    float d;
    asm("v_min_num_f32 %0, %1, %2" : "=v"(d) : "v"(a), "v"(b));
    return d;
}

// ---------------------------------------------------------------------------
// Kernel 1: pack targets y -> (-2y0, -2y1, -2y2, y.y) as float4; zero d_out.
// ---------------------------------------------------------------------------
__global__ void chamfer_pack_kernel(const float* __restrict__ v,
                                    float4* __restrict__ packed,
                                    float* __restrict__ out) {
    int idx = blockIdx.x * blockDim.x + threadIdx.x;
    if (idx < BATCH * MPTS) {
        float y0 = v[idx * 3 + 0];
        float y1 = v[idx * 3 + 1];
        float y2 = v[idx * 3 + 2];
        float4 p;
        p.x = -2.0f * y0;
        p.y = -2.0f * y1;
        p.z = -2.0f * y2;
        p.w = y0 * y0 + y1 * y1 + y2 * y2;
        packed[idx] = p;
    }
    if (idx == 0) out[0] = 0.0f;
}

// ---------------------------------------------------------------------------
// Kernel 2: per wave: hold QT A-tiles (queries), sweep a segment of target
// tiles with V_WMMA_F32_16X16X4_F32. Two tiles per step, folded into the
// running min with one v_min3_num_f32 per element (4 VALU per WMMA).
// s_ij = y.y - 2 x.y  (x.x added in the final reduce).
// ---------------------------------------------------------------------------
__global__ __launch_bounds__(BLOCK_THREADS)
void chamfer_wmma_kernel(const float* __restrict__ vpred,
                         const float2* __restrict__ packed2,   // packed as float2 pairs
                         float* __restrict__ partial) {        // [B*N][MSEG]
    const int lane = threadIdx.x & 31;
    const int w = blockIdx.x * WAVES_PER_BLOCK + (threadIdx.x >> 5);
    const int b = w / WAVES_PER_BATCH;
    const int rem = w % WAVES_PER_BATCH;
    const int qgroup = rem / MSEG;            // which group of QT query tiles
    const int seg = rem % MSEG;               // which target segment
    const int qbase = qgroup * (QT * 16);
    const int jbase = seg * TILES_PER_SEG * 16;   // first target point index

    const int r = lane & 15;
    const int half = lane >> 4;

    // A-matrix tiles: lane r half 0 -> (x0, x1); half 1 -> (x2, 1.0)
    v2f A[QT];
#pragma unroll
    for (int t = 0; t < QT; ++t) {
        const int q = qbase + t * 16 + r;
        const float* xp = vpred + ((size_t)b * NPTS + q) * 3;
        float x0 = xp[0], x1 = xp[1], x2 = xp[2];
        A[t].x = half ? x2 : x0;
        A[t].y = half ? 1.0f : x1;
    }

    v8f mn[QT];
#pragma unroll
    for (int t = 0; t < QT; ++t)
#pragma unroll
        for (int i = 0; i < 8; ++i) mn[t][i] = __builtin_inff();

    // B-matrix fetch: lane r half h loads float2 at point (jbase + r), pair h.
    const float2* bp = packed2 + (((size_t)b * MPTS + jbase + r) * 2 + half);

    for (int j = 0; j < TILES_PER_SEG; j += 2) {
        float2 b0 = bp[(j + 0) * 32];         // 16 points * 16B per tile step
        float2 b1 = bp[(j + 1) * 32];
        v2f B0, B1;
        B0.x = b0.x; B0.y = b0.y;
        B1.x = b1.x; B1.y = b1.y;

        v8f c0[QT], c1[QT];
#pragma unroll
        for (int t = 0; t < QT; ++t) {
            v8f z = {};
            c0[t] = __builtin_amdgcn_wmma_f32_16x16x4_f32(
                        false, A[t], false, B0, (short)0, z, false, false);
        }
#pragma unroll
        for (int t = 0; t < QT; ++t) {
            v8f z = {};
            c1[t] = __builtin_amdgcn_wmma_f32_16x16x4_f32(
                        false, A[t], false, B1, (short)0, z, false, false);
        }
#pragma unroll
        for (int t = 0; t < QT; ++t)
#pragma unroll
            for (int i = 0; i < 8; ++i)
                mn[t][i] = min3f(mn[t][i], c0[t][i], c1[t][i]);
    }

    // Min across the 16 columns (lanes within each half), then write partials.
#pragma unroll
    for (int t = 0; t < QT; ++t) {
#pragma unroll
        for (int i = 0; i < 8; ++i) {
            float m = mn[t][i];
            m = min2f(m, __shfl_xor(m, 1, 32));
            m = min2f(m, __shfl_xor(m, 2, 32));
            m = min2f(m, __shfl_xor(m, 4, 32));
            m = min2f(m, __shfl_xor(m, 8, 32));
            if (r == 0) {
                const int q = qbase + t * 16 + i + 8 * half;  // row m = i + 8*half
                partial[((size_t)b * NPTS + q) * MSEG + seg] = m;
            }
        }
    }
}

// ---------------------------------------------------------------------------
// Kernel 3: min over segments, add x.x, scale, tree-reduce, atomicAdd scalar.
// ---------------------------------------------------------------------------
__global__ __launch_bounds__(BLOCK_THREADS)
void chamfer_reduce_kernel(const float* __restrict__ vpred,
                           const float* __restrict__ partial,
                           float* __restrict__ out) {
    const int idx = blockIdx.x * blockDim.x + threadIdx.x;  // over B*N points
    const float* p = partial + (size_t)idx * MSEG;
    float m = min3f(p[0], p[1], p[2]);
    m = min3f(m, p[3], p[4]);
    m = min3f(m, p[5], p[6]);
    m = min2f(m, p[7]);
    const float* xp = vpred + (size_t)idx * 3;
    float x0 = xp[0], x1 = xp[1], x2 = xp[2];
    float val = (m + x0 * x0 + x1 * x1 + x2 * x2) *
                (1.0f / ((float)BATCH * (float)NPTS));
#pragma unroll
    for (int off = 1; off < 32; off <<= 1)
        val += __shfl_xor(val, off, 32);
    if ((threadIdx.x & 31) == 0) atomicAdd(out, val);
}

// ---------------------------------------------------------------------------
extern "C" void kernel_launch(void* const* d_in, const int* in_sizes, int n_in,
                              void* d_out, int out_size, void* d_ws, size_t ws_size,
                              hipStream_t stream) {
    const float* v = (const float*)d_in[0];       // targets  [B, M, 3]
    const float* v_pred = (const float*)d_in[1];  // queries  [B, N, 3]
    float* out = (float*)d_out;                   // scalar loss

    // Workspace layout: packed targets (B*M float4) | partial mins (B*N*MSEG f32)
    float4* packed = (float4*)d_ws;
    float* partial = (float*)((char*)d_ws + (size_t)BATCH * MPTS * sizeof(float4));

    {
        int n = BATCH * MPTS;
        chamfer_pack_kernel<<<(n + BLOCK_THREADS - 1) / BLOCK_THREADS,
                              BLOCK_THREADS, 0, stream>>>(v, packed, out);
    }
    chamfer_wmma_kernel<<<MAIN_BLOCKS, BLOCK_THREADS, 0, stream>>>(
        v_pred, (const float2*)packed, partial);
    {
        int n = BATCH * NPTS;
        chamfer_reduce_kernel<<<n / BLOCK_THREADS, BLOCK_THREADS, 0, stream>>>(
            v_pred, partial, out);
    }
}